// MambaG2G_6536940224928
// MI455X (gfx1250) — compile-verified
//
#include <hip/hip_runtime.h>
#include <math.h>
#include <stdint.h>

// ---------------------------------------------------------------------------
// Model dims
// ---------------------------------------------------------------------------
#define TT   8
#define NN   2048
#define DV   256
#define DI   512
#define DS_  32
#define DC_  4
#define DTR_ 16
#define DO_  64
#define NL_  2

typedef __attribute__((ext_vector_type(2))) float v2f;
typedef __attribute__((ext_vector_type(4))) float v4f;
typedef __attribute__((ext_vector_type(8))) float v8f;
typedef __attribute__((ext_vector_type(4))) unsigned int v4u;
typedef __attribute__((ext_vector_type(4))) int v4i;
typedef __attribute__((ext_vector_type(8))) int v8i;

// ---------------------------------------------------------------------------
// Generic fp32 WMMA GEMM:  C[M,N] = act( A[M,K] @ op(B) + bias )  (opt. +=C)
//   BTRANS=true : B stored [N,K] row-major (compute A @ B^T, weight form)
//   BTRANS=false: B stored [K,N] row-major
// Block tile 64x64, K-step 16, 128 threads = 4 waves (2x2), wave tile 32x32.
// A-tiles are DMA'd HBM->LDS by the Tensor Data Mover (tensor_load_to_lds,
// TENSORcnt), double-buffered so one DMA is always in flight behind the
// V_WMMA_F32_16X16X4_F32 inner loop. TDM pad feature reproduces the 17-float
// LDS row pitch (conflict-free fragment reads).
// NOTE: every call in this model has M % 64 == 0 and K % 16 == 0, so A-tiles
// are always fully in-bounds for the DMA.
// ---------------------------------------------------------------------------
template <bool BTRANS>
__global__ __launch_bounds__(128)
void gemm_f32_wmma(const float* __restrict__ A, int lda,
                   const float* __restrict__ B, int ldb,
                   const float* __restrict__ bias, int bias_mode,  // 0 none, 1 per-col, 2 per-row
                   float* __restrict__ C, int ldc,
                   int M, int Nn, int K,
                   int act,       // 0 none, 1 softplus, 2 elu(tanh(x)), 3 elu(x)+1+1e-14
                   int accum)     // 0: C = v,  1: C += v
{
    constexpr int BM = 64, BN = 64, BK = 16;
    constexpr int AP = BK + 1;                 // LDS pad: gcd(17,64)=1 -> conflict-free
    __shared__ float As[2][BM * AP];
    __shared__ float Bs[2][BK * BN];

    const int tid  = threadIdx.x;
    const int lane = tid & 31;
    const int w    = tid >> 5;                 // wave 0..3
    const int wr   = (w >> 1) * 32;            // wave row offset in tile
    const int wc   = (w & 1) * 32;             // wave col offset in tile
    const int lh   = (lane < 16) ? 0 : 1;      // lane half (ISA 7.12.2)
    const int ll   = lane & 15;

    const int bm0 = blockIdx.y * BM;
    const int bn0 = blockIdx.x * BN;

    // ---- TDM issue: DMA a 64x16 fp32 A-tile into As[buf] (wave 0 only) ----
    auto issueA = [&](int buf, int k0) {
        if (tid < 32) {
            unsigned long long ga = (unsigned long long)(uintptr_t)A +
                                    4ull * ((unsigned long long)bm0 * (unsigned)lda + (unsigned)k0);
            unsigned lds = (unsigned)(uintptr_t)(void*)&As[buf][0];
            v4u g0;
            g0[0] = 1u;                                        // count=1 (valid D#)
            g0[1] = lds;                                       // lds_addr
            g0[2] = (unsigned)ga;                              // global_addr[31:0]
            g0[3] = (unsigned)((ga >> 32) & 0x1FFFFFFull)      // global_addr[56:32]
                    | (2u << 30);                              // type=2 ("image")
            v8i g1;
            g1[0] = (int)((2u << 16)        // data_size = 4 bytes
                        | (1u << 20)        // pad_enable
                        | (3u << 22));      // pad_interval: every 16 DWORDs (+pad_amount=0 -> 1 DWORD)
            g1[1] = (int)(16u << 16);       // tensor_dim0 = 16 (bits 79:48, low half)
            g1[2] = (int)(64u << 16);       // tensor_dim0 hi | tensor_dim1 = 64 (low half)
            g1[3] = (int)(16u << 16);       // tensor_dim1 hi | tile_dim0 = 16
            g1[4] = 64;                     // tile_dim1 = 64, tile_dim2 = 0
            g1[5] = lda;                    // tensor_dim0_stride (element units), low 32
            g1[6] = 0;
            g1[7] = 0;
            v4i z4 = {0, 0, 0, 0};
            v8i z8 = {0, 0, 0, 0, 0, 0, 0, 0};
            // amdgpu-toolchain (clang-23) 6-arg form:
            // (g0 v4u, g1 v8i, g2 v4i, g3 v4i, v8i, cpol)
            __builtin_amdgcn_tensor_load_to_lds(g0, g1, z4, z4, z8, 0);
        }
    };

    // ---- manual (transposing) B-tile stage into Bs[buf][k][n] ------------
    auto loadB = [&](int buf, int k0) {
        if constexpr (BTRANS) {
            #pragma unroll
            for (int i = 0; i < 2; ++i) {
                int linear = tid + i * 128;
                int n  = linear >> 2;
                int kv = (linear & 3) * 4;
                int gn = bn0 + n;
                v4f val = {0.f, 0.f, 0.f, 0.f};
                if (gn < Nn) {
                    const float* bp = B + (long)gn * ldb + k0 + kv;
                    val = *(const v4f*)bp;
                    if (k0 + BK < K) __builtin_prefetch(bp + BK, 0, 1);  // global_prefetch_b8
                }
                Bs[buf][(kv + 0) * BN + n] = val[0];
                Bs[buf][(kv + 1) * BN + n] = val[1];
                Bs[buf][(kv + 2) * BN + n] = val[2];
                Bs[buf][(kv + 3) * BN + n] = val[3];
            }
        } else {
            #pragma unroll
            for (int i = 0; i < 2; ++i) {
                int linear = tid + i * 128;
                int k  = linear >> 4;          // 16 float4 per 64-wide row
                int nv = (linear & 15) * 4;
                int gn = bn0 + nv;
                v4f val = {0.f, 0.f, 0.f, 0.f};
                if (gn + 3 < Nn) {
                    val = *(const v4f*)(B + (long)(k0 + k) * ldb + gn);
                } else {
                    if (gn + 0 < Nn) val[0] = B[(long)(k0 + k) * ldb + gn + 0];
                    if (gn + 1 < Nn) val[1] = B[(long)(k0 + k) * ldb + gn + 1];
                    if (gn + 2 < Nn) val[2] = B[(long)(k0 + k) * ldb + gn + 2];
                }
                *(v4f*)&Bs[buf][k * BN + nv] = val;
            }
        }
    };

    v8f acc[2][2] = {};
    const int ntiles = K >> 4;

    // prologue: start tile 0
    issueA(0, 0);
    loadB(0, 0);

    for (int i = 0; i < ntiles; ++i) {
        const int cur = i & 1;
        if (i + 1 < ntiles) {
            issueA(cur ^ 1, (i + 1) << 4);     // keep one DMA in flight
            loadB(cur ^ 1, (i + 1) << 4);
            if (tid < 32) __builtin_amdgcn_s_wait_tensorcnt((short)1);  // tile `cur` done
        } else {
            if (tid < 32) __builtin_amdgcn_s_wait_tensorcnt((short)0);
        }
        __syncthreads();

        // ---- 4 k-steps of 4, 2x2 fragments per wave ---------------------
        #pragma unroll
        for (int kk = 0; kk < BK; kk += 4) {
            v2f af[2], bf[2];
            #pragma unroll
            for (int fi = 0; fi < 2; ++fi) {
                const float* p = &As[cur][(wr + fi * 16 + ll) * AP + kk + lh * 2];
                af[fi][0] = p[0];
                af[fi][1] = p[1];
            }
            #pragma unroll
            for (int fj = 0; fj < 2; ++fj) {
                int n = wc + fj * 16 + ll;
                int kr = kk + lh * 2;
                bf[fj][0] = Bs[cur][kr * BN + n];
                bf[fj][1] = Bs[cur][(kr + 1) * BN + n];
            }
            #pragma unroll
            for (int fi = 0; fi < 2; ++fi)
                #pragma unroll
                for (int fj = 0; fj < 2; ++fj)
                    acc[fi][fj] = __builtin_amdgcn_wmma_f32_16x16x4_f32(
                        false, af[fi], false, bf[fj], (short)0, acc[fi][fj],
                        false, false);
        }
        __syncthreads();
    }

    // ---- epilogue: ISA C/D layout (reg=row, lane=col, halves split M) ----
    #pragma unroll
    for (int fi = 0; fi < 2; ++fi) {
        int mbase = bm0 + wr + fi * 16 + lh * 8;
        #pragma unroll
        for (int fj = 0; fj < 2; ++fj) {
            int gn = bn0 + wc + fj * 16 + ll;
            if (gn >= Nn) continue;
            #pragma unroll
            for (int r = 0; r < 8; ++r) {
                int gm = mbase + r;
                if (gm >= M) continue;
                float v = acc[fi][fj][r];
                if (bias_mode == 1)      v += bias[gn];
                else if (bias_mode == 2) v += bias[gm];
                if (act == 1) {                         // softplus
                    v = (v > 20.f) ? v : log1pf(expf(v));
                } else if (act == 2) {                  // elu(tanh(x))
                    float t = tanhf(v);
                    v = (t > 0.f) ? t : expm1f(t);
                } else if (act == 3) {                  // elu(x)+1+1e-14
                    v = ((v > 0.f) ? v : expm1f(v)) + 1.0f + 1e-14f;
                }
                float* cp = C + (long)gm * ldc + gn;
                if (accum) *cp += v; else *cp = v;
            }
        }
    }
}

// ---------------------------------------------------------------------------
// RMSNorm over rows of a (2048 x 256) matrix
// ---------------------------------------------------------------------------
__global__ __launch_bounds__(DV)
void rmsnorm_kernel(const float* __restrict__ z, const float* __restrict__ w,
                    float* __restrict__ xn)
{
    __shared__ float red[DV];
    int m = blockIdx.x, tid = threadIdx.x;
    float v = z[m * DV + tid];
    red[tid] = v * v;
    __syncthreads();
    for (int s = DV / 2; s > 0; s >>= 1) {
        if (tid < s) red[tid] += red[tid + s];
        __syncthreads();
    }
    float scale = rsqrtf(red[0] * (1.0f / DV) + 1e-5f);
    xn[m * DV + tid] = v * scale * w[tid];
}

// ---------------------------------------------------------------------------
// Depthwise causal conv (width 4) along t + SiLU.  xi = xz[:, 0:512]
// ---------------------------------------------------------------------------
__global__ __launch_bounds__(256)
void conv_silu_kernel(const float* __restrict__ xz, const float* __restrict__ cw,
                      const float* __restrict__ cb, float* __restrict__ xc)
{
    int idx = blockIdx.x * 256 + threadIdx.x;          // over 2048*512
    if (idx >= 2048 * DI) return;
    int d = idx & (DI - 1);
    int m = idx >> 9;                                   // a*8+t
    int t = m & 7, a = m >> 3;
    float s = cb[d];
    #pragma unroll
    for (int k = 0; k < DC_; ++k) {
        int tt = t + k - (DC_ - 1);
        if (tt >= 0) s += xz[((a << 3) + tt) * (2 * DI) + d] * cw[d * DC_ + k];
    }
    float sig = 1.f / (1.f + expf(-s));
    xc[idx] = s * sig;
}

// ---------------------------------------------------------------------------
// Fused selective scan: thread per (a, d); 32-element SSM state in registers;
// B/C rows for this 'a' staged in LDS; fuses (y + D*x) * silu(z_gate).
// ---------------------------------------------------------------------------
__global__ __launch_bounds__(256)
void scan_kernel(const float* __restrict__ delta, const float* __restrict__ xc,
                 const float* __restrict__ dbc,  const float* __restrict__ xz,
                 const float* __restrict__ A_log, const float* __restrict__ Dp,
                 float* __restrict__ ybuf)
{
    __shared__ float Bm[TT][DS_];
    __shared__ float Cm[TT][DS_];
    int a   = blockIdx.x;
    int tid = threadIdx.x;
    int d   = blockIdx.y * 256 + tid;
    {
        int t = tid >> 5, s = tid & 31;
        const float* row = dbc + (long)(a * TT + t) * (DTR_ + 2 * DS_);
        Bm[t][s] = row[DTR_ + s];
        Cm[t][s] = row[DTR_ + DS_ + s];
    }
    __syncthreads();

    float Ad[DS_], h[DS_];
    #pragma unroll
    for (int s = 0; s < DS_; ++s) { Ad[s] = -expf(A_log[d * DS_ + s]); h[s] = 0.f; }
    float dcoef = Dp[d];

    for (int t = 0; t < TT; ++t) {
        int m = a * TT + t;
        float dl = delta[m * DI + d];
        float xv = xc[m * DI + d];
        float zg = xz[m * (2 * DI) + DI + d];
        float y = 0.f;
        #pragma unroll
        for (int s = 0; s < DS_; ++s) {
            float dA = expf(dl * Ad[s]);
            h[s] = dA * h[s] + dl * Bm[t][s] * xv;
            y += h[s] * Cm[t][s];
        }
        float sig = 1.f / (1.f + expf(-zg));
        ybuf[m * DI + d] = (y + dcoef * xv) * (zg * sig);
    }
}

// ---------------------------------------------------------------------------
// e[a][b] = mean over t of z[a][t][b]
// ---------------------------------------------------------------------------
__global__ __launch_bounds__(256)
void mean_kernel(const float* __restrict__ z, float* __restrict__ e)
{
    int idx = blockIdx.x * 256 + threadIdx.x;          // 256*256
    int b = idx & (DV - 1), a = idx >> 8;
    float s = 0.f;
    #pragma unroll
    for (int t = 0; t < TT; ++t) s += z[((a << 3) + t) * DV + b];
    e[idx] = s * (1.0f / TT);
}

// ---------------------------------------------------------------------------
static inline void launch_gemm(bool btrans,
                               const float* A, int lda, const float* B, int ldb,
                               const float* bias, int bias_mode,
                               float* C, int ldc, int M, int Nn, int K,
                               int act, int accum, hipStream_t st)
{
    dim3 grid((Nn + 63) / 64, (M + 63) / 64, 1);
    if (btrans)
        gemm_f32_wmma<true><<<grid, 128, 0, st>>>(A, lda, B, ldb, bias, bias_mode,
                                                  C, ldc, M, Nn, K, act, accum);
    else
        gemm_f32_wmma<false><<<grid, 128, 0, st>>>(A, lda, B, ldb, bias, bias_mode,
                                                   C, ldc, M, Nn, K, act, accum);
}

extern "C" void kernel_launch(void* const* d_in, const int* in_sizes, int n_in,
                              void* d_out, int out_size, void* d_ws, size_t ws_size,
                              hipStream_t stream)
{
    (void)in_sizes; (void)n_in; (void)out_size; (void)ws_size;

    const float* inp       = (const float*)d_in[0];
    const float* enc_W     = (const float*)d_in[1];
    const float* enc_b     = (const float*)d_in[2];
    const float* red_W     = (const float*)d_in[3];
    const float* red_b     = (const float*)d_in[4];
    const float* norm_w    = (const float*)d_in[5];
    const float* in_proj_W = (const float*)d_in[6];
    const float* conv_w    = (const float*)d_in[7];
    const float* conv_b    = (const float*)d_in[8];
    const float* xproj_W   = (const float*)d_in[9];
    const float* dtproj_W  = (const float*)d_in[10];
    const float* dtproj_b  = (const float*)d_in[11];
    const float* A_log     = (const float*)d_in[12];
    const float* D_p       = (const float*)d_in[13];
    const float* outproj_W = (const float*)d_in[14];
    const float* out_fc_W  = (const float*)d_in[15];
    const float* out_fc_b  = (const float*)d_in[16];
    const float* mu_W      = (const float*)d_in[17];
    const float* mu_b      = (const float*)d_in[18];
    const float* sig_W     = (const float*)d_in[19];
    const float* sig_b     = (const float*)d_in[20];

    // workspace carve-up (floats)
    float* ws    = (float*)d_ws;
    float* O1    = ws;                       // 2048*256   (per-t reuse)
    float* z     = O1    + 2048 * DV;        // 2048*256   rows m=a*8+t
    float* xn    = z     + 2048 * DV;        // 2048*256
    float* xz    = xn    + 2048 * DV;        // 2048*1024
    float* xc    = xz    + 2048 * 2 * DI;    // 2048*512
    float* dbc   = xc    + 2048 * DI;        // 2048*80
    float* delta = dbc   + 2048 * (DTR_ + 2 * DS_); // 2048*512
    float* ybuf  = delta + 2048 * DI;        // 2048*512
    float* e     = ybuf  + 2048 * DI;        // 256*256

    // Phase A+B: per t, O1 = inp[t] @ enc_W^T + enc_b ; z[:,t,:] = red_W @ O1 + red_b
    for (int t = 0; t < TT; ++t) {
        launch_gemm(true,  inp + (long)t * NN * NN, NN, enc_W, NN,
                    enc_b, 1, O1, DV, NN, DV, NN, 0, 0, stream);
        launch_gemm(false, red_W, NN, O1, DV,
                    red_b, 2, z + t * DV, TT * DV, DV, DV, NN, 0, 0, stream);
    }

    // Mamba layers
    for (int l = 0; l < NL_; ++l) {
        rmsnorm_kernel<<<2048, DV, 0, stream>>>(z, norm_w + l * DV, xn);
        launch_gemm(true, xn, DV, in_proj_W + (long)l * 2 * DI * DV, DV,
                    nullptr, 0, xz, 2 * DI, 2048, 2 * DI, DV, 0, 0, stream);
        conv_silu_kernel<<<(2048 * DI + 255) / 256, 256, 0, stream>>>(
            xz, conv_w + l * DI * DC_, conv_b + l * DI, xc);
        launch_gemm(true, xc, DI, xproj_W + (long)l * (DTR_ + 2 * DS_) * DI, DI,
                    nullptr, 0, dbc, DTR_ + 2 * DS_, 2048, DTR_ + 2 * DS_, DI,
                    0, 0, stream);
        launch_gemm(true, dbc, DTR_ + 2 * DS_, dtproj_W + (long)l * DI * DTR_, DTR_,
                    dtproj_b + l * DI, 1, delta, DI, 2048, DI, DTR_,
                    /*softplus*/1, 0, stream);
        scan_kernel<<<dim3(DV, DI / 256), 256, 0, stream>>>(
            delta, xc, dbc, xz, A_log + (long)l * DI * DS_, D_p + l * DI, ybuf);
        launch_gemm(true, ybuf, DI, outproj_W + (long)l * DV * DI, DI,
                    nullptr, 0, z, DV, 2048, DV, DI, 0, /*accum*/1, stream);
    }

    // Head: e = mean_t z ; X[n,a] = elu(tanh(out_fc_W @ e^T + b_row)) ; mu/sigma
    mean_kernel<<<(DV * DV) / 256, 256, 0, stream>>>(z, e);
    float* Xout = (float*)d_out;
    launch_gemm(true, out_fc_W, DV, e, DV,
                out_fc_b, 2, Xout, DV, NN, DV, DV, /*elu(tanh)*/2, 0, stream);
    launch_gemm(true, Xout, DV, mu_W, DV,
                mu_b, 1, Xout + (long)NN * DV, DO_, NN, DO_, DV, 0, 0, stream);
    launch_gemm(true, Xout, DV, sig_W, DV,
                sig_b, 1, Xout + (long)NN * DV + (long)NN * DO_, DO_, NN, DO_, DV,
                /*elu+1*/3, 0, stream);
}